// Model_52192442581135
// MI455X (gfx1250) — compile-verified
//
#include <hip/hip_runtime.h>

typedef __attribute__((ext_vector_type(2))) float v2f;
typedef __attribute__((ext_vector_type(8))) float v8f;

#define FT_OUT 257
#define HALF   256
#define NACT   32
#define ROWS   16    // batch rows per block
#define THREADS 256  // 8 waves (wave32)

__global__ __launch_bounds__(THREADS)
void nnue_fused(const int* __restrict__ wft, const int* __restrict__ bft,
                const float* __restrict__ stm, const float* __restrict__ W,
                const float* __restrict__ ftb, const float* __restrict__ fcw,
                const float* __restrict__ fcb, float* __restrict__ out, int Bn)
{
    // 516 stride: 516 % 64 == 4 -> conflict-free column reads in WMMA phase
    __shared__ float lds_x[ROWS][516];
    __shared__ float lds_fcw[512];
    __shared__ float lds_ps[ROWS];
    __shared__ float lds_part[8][ROWS];

    const int tid = threadIdx.x;
    const int g   = tid >> 4;   // row group 0..15 within block
    const int j   = tid & 15;   // lane within row group
    int row = blockIdx.x * ROWS + g;
    if (row > Bn - 1) row = Bn - 1;   // safe tail (B is a multiple of 16 anyway)

    // Stage fc_w into LDS (broadcast-read later by all 8 waves)
    lds_fcw[tid]       = fcw[tid];
    lds_fcw[tid + 256] = fcw[tid + 256];

    // ---------------- Phase 1: gather + accumulate (L2-resident table) -----
    float wacc[16], bacc[16];
#pragma unroll
    for (int i = 0; i < 16; ++i) {
        float bv = ftb[j + 16 * i];
        wacc[i] = bv; bacc[i] = bv;
    }
    float wps = (j == 0) ? ftb[256] : 0.f;
    float bps = wps;

    const long ibase = (long)row * NACT;
    for (int k = 0; k < NACT; ++k) {
        int wi = wft[ibase + k];
        if (wi >= 0) {
            const float* r = W + (long)wi * FT_OUT;
#pragma unroll
            for (int i = 0; i < 16; ++i) wacc[i] += r[j + 16 * i];
            if (j == 0) wps += r[256];
        }
        int bi = bft[ibase + k];
        if (bi >= 0) {
            const float* r = W + (long)bi * FT_OUT;
#pragma unroll
            for (int i = 0; i < 16; ++i) bacc[i] += r[j + 16 * i];
            if (j == 0) bps += r[256];
        }
    }

    // ---------------- Phase 2: stm blend + clip -> LDS ---------------------
    const float s = stm[row];
#pragma unroll
    for (int i = 0; i < 16; ++i) {
        int c = j + 16 * i;
        float w = wacc[i], b = bacc[i];
        float lo = fminf(fmaxf((1.f - s) * w + s * b, 0.f), 1.f);
        float hi = fminf(fmaxf((1.f - s) * b + s * w, 0.f), 1.f);
        lds_x[g][c]        = lo;
        lds_x[g][HALF + c] = hi;
    }
    if (j == 0) lds_ps[g] = (wps - bps) * (0.5f - s);
    __syncthreads();

    // ---------------- Phase 3: FC via V_WMMA_F32_16X16X4_F32 ---------------
    // D(16x16) = sum_K A(16x4) * B(4x16); B has fc_w in column N=0, zeros else
    // (branchless: 0/1 lane mask times LDS broadcast). Each wave handles a
    // K-slice of 64 as two independent 8-deep accumulation chains.
    const int lane  = tid & 31;
    const int wv    = tid >> 5;          // wave id 0..7
    const int mn    = lane & 15;         // A: row M ; B: col N
    const int kk    = (lane >> 4) * 2;   // K sub-offset per ISA f32 A/B layout
    const float bmask = (mn == 0) ? 1.f : 0.f;
    v8f acc0 = {}, acc1 = {};
#pragma unroll
    for (int t = 0; t < 8; ++t) {
        int kb0 = wv * 64 + t * 8 + kk;      // chain 0: k = 0,8,16,...
        int kb1 = kb0 + 4;                    // chain 1: k = 4,12,20,...
        v2f a0, b0, a1, b1;
        a0.x = lds_x[mn][kb0];     a0.y = lds_x[mn][kb0 + 1];
        b0.x = bmask * lds_fcw[kb0];
        b0.y = bmask * lds_fcw[kb0 + 1];
        a1.x = lds_x[mn][kb1];     a1.y = lds_x[mn][kb1 + 1];
        b1.x = bmask * lds_fcw[kb1];
        b1.y = bmask * lds_fcw[kb1 + 1];
        acc0 = __builtin_amdgcn_wmma_f32_16x16x4_f32(
            false, a0, false, b0, (short)0, acc0, false, false);
        acc1 = __builtin_amdgcn_wmma_f32_16x16x4_f32(
            false, a1, false, b1, (short)0, acc1, false, false);
    }
    // Column N=0 of D lives in lane 0 (M=0..7 in v0..v7) and lane 16 (M=8..15).
    if (mn == 0) {
        int mb = (lane >> 4) * 8;
#pragma unroll
        for (int v = 0; v < 8; ++v) lds_part[wv][mb + v] = acc0[v] + acc1[v];
    }
    __syncthreads();

    // ---------------- Phase 4: reduce 8 wave-partials + bias + psqt --------
    if (tid < ROWS) {
        float sum = fcb[0] + lds_ps[tid];
#pragma unroll
        for (int w = 0; w < 8; ++w) sum += lds_part[w][tid];
        int orow = blockIdx.x * ROWS + tid;
        if (orow < Bn) out[orow] = sum;
    }
}

extern "C" void kernel_launch(void* const* d_in, const int* in_sizes, int n_in,
                              void* d_out, int out_size, void* d_ws, size_t ws_size,
                              hipStream_t stream) {
    (void)n_in; (void)d_ws; (void)ws_size; (void)out_size;
    const int*   wft = (const int*)d_in[0];    // (B, 32) int32
    const int*   bft = (const int*)d_in[1];    // (B, 32) int32
    const float* stm = (const float*)d_in[2];  // (B, 1)  f32
    const float* W   = (const float*)d_in[3];  // (40960, 257) f32
    const float* ftb = (const float*)d_in[4];  // (257,)  f32
    const float* fcw = (const float*)d_in[5];  // (1, 512) f32
    const float* fcb = (const float*)d_in[6];  // (1,)    f32
    float* out = (float*)d_out;                // (B, 1)  f32

    const int Bn = in_sizes[0] / NACT;         // 16384
    const int grid = (Bn + ROWS - 1) / ROWS;   // 1024 blocks
    nnue_fused<<<grid, THREADS, 0, stream>>>(wft, bft, stm, W, ftb, fcw, fcb, out, Bn);
}